// ConditionalSDFModel_30107720745798
// MI455X (gfx1250) — compile-verified
//
#include <hip/hip_runtime.h>

// ---------------------------------------------------------------------------
// Conditional-SDF MoE forward for MI455X (gfx1250, wave32, WMMA).
//   N=65536 points, E=8 experts, D=3, L=64, H=256.
//   Per point: h0=relu([x,emb[l]]@W0+b0); h1=relu(h0@W1+b1);
//              h2=relu(h1@W2+b2); out=h2@W3+b3  -- expert selected by label.
// Strategy: group points by expert, tile 64 rows per block, fp32 WMMA
// (v_wmma_f32_16x16x4_f32) for the two 256x256 layers, layer-0 collapsed via
// the shared-latent rank structure, layer-3 as a VALU dot reduction.
// ---------------------------------------------------------------------------

#define NPTS 65536
#define NE   8
#define ND   3
#define NL   64
#define NH   256
#define NIN  (ND + NL)   // 67
#define TM   64          // rows (points) per block
#define LDH  268         // padded LDS row stride in floats (268 mod 64 = 12 -> conflict-free)

// meta layout in workspace (ints)
#define M_COUNTS   0     // 8
#define M_OFFSETS  8     // 9
#define M_CURSOR   17    // 8
#define M_TILEBASE 25    // 9
#define M_SIZE     64

typedef __attribute__((ext_vector_type(2))) float v2f;
typedef __attribute__((ext_vector_type(8))) float v8f;

// ------------------------------ prep kernels -------------------------------

__global__ void zero_meta_kernel(int* __restrict__ meta) {
  if (threadIdx.x < M_SIZE) meta[threadIdx.x] = 0;
}

__global__ void count_kernel(const int* __restrict__ labels, int* __restrict__ meta) {
  const int i = blockIdx.x * blockDim.x + threadIdx.x;
  if (i < NPTS) atomicAdd(&meta[M_COUNTS + labels[i]], 1);
}

__global__ void scan_kernel(int* __restrict__ meta) {
  if (threadIdx.x == 0 && blockIdx.x == 0) {
    int off = 0;
    for (int e = 0; e < NE; ++e) {
      meta[M_OFFSETS + e] = off;
      meta[M_CURSOR + e]  = off;
      off += meta[M_COUNTS + e];
    }
    meta[M_OFFSETS + NE] = off;
    int tb = 0;
    for (int e = 0; e < NE; ++e) {
      meta[M_TILEBASE + e] = tb;
      tb += (meta[M_COUNTS + e] + TM - 1) / TM;
    }
    meta[M_TILEBASE + NE] = tb;
  }
}

__global__ void scatter_kernel(const int* __restrict__ labels,
                               int* __restrict__ meta, int* __restrict__ perm) {
  const int i = blockIdx.x * blockDim.x + threadIdx.x;
  if (i < NPTS) {
    const int p = atomicAdd(&meta[M_CURSOR + labels[i]], 1);
    perm[p] = i;
  }
}

// c0[e][h] = b0[e][h] + sum_k emb[e][k] * W0[e][3+k][h]
// (latent rows of layer 0 are identical for every point of expert e)
__global__ void c0_kernel(const float* __restrict__ emb, const float* __restrict__ W0,
                          const float* __restrict__ b0, float* __restrict__ c0) {
  const int e = blockIdx.x;
  const int h = threadIdx.x;
  const float* W0e = W0 + (size_t)e * NIN * NH;
  float s = b0[e * NH + h];
#pragma unroll 8
  for (int k = 0; k < NL; ++k)
    s = fmaf(emb[e * NL + k], W0e[(size_t)(ND + k) * NH + h], s);
  c0[e * NH + h] = s;
}

// ------------------------------ WMMA GEMM ----------------------------------
// C[64x256] = relu(A[64x256] @ Wg[256x256] + bg), A/C in LDS (stride LDH),
// Wg/bg in global (L2-resident). 8 waves: wave w owns columns [32w, 32w+32).
// Fragment layouts per CDNA5 ISA 7.12.2:
//   A 16x4 f32 : lane<16 -> {K0,K1}, lane>=16 -> {K2,K3}, M = lane&15
//   B 4x16 f32 : lane<16 -> rows {K0,K1}, lane>=16 -> rows {K2,K3}, N = lane&15
//   C/D 16x16  : VGPR r -> M = r + (lane<16 ? 0 : 8), N = lane&15
__device__ __forceinline__ void gemm_relu(const float* __restrict__ A,
                                          float* __restrict__ C,
                                          const float* __restrict__ Wg,
                                          const float* __restrict__ bg) {
  const int lane  = threadIdx.x & 31;
  const int wave  = threadIdx.x >> 5;
  const int n0    = wave * 32;
  const int mlane = lane & 15;
  const int khalf = lane >> 4;              // 0 -> K pair {0,1}; 1 -> {2,3}

  const float bias0 = bg[n0 + mlane];
  const float bias1 = bg[n0 + 16 + mlane];

  v8f acc[4][2];
#pragma unroll
  for (int rt = 0; rt < 4; ++rt)
#pragma unroll
    for (int k = 0; k < 8; ++k) {
      acc[rt][0][k] = bias0;
      acc[rt][1][k] = bias1;
    }

#pragma unroll 4
  for (int k0 = 0; k0 < NH; k0 += 4) {
    const int kk = k0 + (khalf << 1);
    v2f bf0, bf1;
    bf0.x = Wg[(size_t)kk * NH + n0 + mlane];
    bf0.y = Wg[(size_t)(kk + 1) * NH + n0 + mlane];
    bf1.x = Wg[(size_t)kk * NH + n0 + 16 + mlane];
    bf1.y = Wg[(size_t)(kk + 1) * NH + n0 + 16 + mlane];
    if (k0 + 8 < NH)  // warm the next B K-slice into the WGP-level cache:
                      // consecutive tiles of the same expert re-read this data.
      __builtin_prefetch(&Wg[(size_t)(k0 + 8) * NH + n0 + mlane], 0, 3);
#pragma unroll
    for (int rt = 0; rt < 4; ++rt) {
      const int m = rt * 16 + mlane;
      v2f a;
      a.x = A[m * LDH + kk];
      a.y = A[m * LDH + kk + 1];
      acc[rt][0] = __builtin_amdgcn_wmma_f32_16x16x4_f32(
          false, a, false, bf0, (short)0, acc[rt][0], false, false);
      acc[rt][1] = __builtin_amdgcn_wmma_f32_16x16x4_f32(
          false, a, false, bf1, (short)0, acc[rt][1], false, false);
    }
  }

#pragma unroll
  for (int rt = 0; rt < 4; ++rt)
#pragma unroll
    for (int nt = 0; nt < 2; ++nt) {
      const int col = n0 + nt * 16 + mlane;
#pragma unroll
      for (int k = 0; k < 8; ++k) {
        const int row = rt * 16 + khalf * 8 + k;
        C[row * LDH + col] = fmaxf(acc[rt][nt][k], 0.0f);
      }
    }
}

// ------------------------------ main kernel --------------------------------

#define SMEM_BYTES ((2 * TM * LDH + TM * 4 + NH + TM * 4) * 4 + TM * 4)

__global__ __launch_bounds__(256) void moe_forward_kernel(
    const float* __restrict__ x,  const float* __restrict__ W0,
    const float* __restrict__ W1, const float* __restrict__ b1,
    const float* __restrict__ W2, const float* __restrict__ b2,
    const float* __restrict__ W3, const float* __restrict__ b3,
    const int* __restrict__ meta, const int* __restrict__ perm,
    const float* __restrict__ c0, float* __restrict__ out) {
  extern __shared__ float smem[];
  float* hA   = smem;                 // TM * LDH
  float* hB   = hA + TM * LDH;        // TM * LDH
  float* xs   = hB + TM * LDH;        // TM * 4
  float* w3s  = xs + TM * 4;          // NH
  float* red  = w3s + NH;             // TM * 4
  int*   ridx = (int*)(red + TM * 4); // TM

  const int b = blockIdx.x;
  if (b >= meta[M_TILEBASE + NE]) return;
  int e = 0;
  while (b >= meta[M_TILEBASE + e + 1]) ++e;
  const int tile  = b - meta[M_TILEBASE + e];
  const int cnt   = meta[M_COUNTS + e];
  const int start = meta[M_OFFSETS + e] + tile * TM;
  const int nrows = min(TM, cnt - tile * TM);

  const int t = threadIdx.x;
  if (t < TM) {
    if (t < nrows) {
      const int i = perm[start + t];
      ridx[t] = i;
      xs[t * 4 + 0] = x[i * 3 + 0];
      xs[t * 4 + 1] = x[i * 3 + 1];
      xs[t * 4 + 2] = x[i * 3 + 2];
    } else {
      ridx[t] = -1;
      xs[t * 4 + 0] = 0.0f; xs[t * 4 + 1] = 0.0f; xs[t * 4 + 2] = 0.0f;
    }
  }
  w3s[t] = W3[e * NH + t];            // blockDim.x == NH
  __syncthreads();

  // ---- layer 0 (collapsed): h0 = relu(x0*W0r0 + x1*W0r1 + x2*W0r2 + c0[e])
  {
    const float* W0e = W0 + (size_t)e * NIN * NH;
    const float w00 = W0e[0 * NH + t];
    const float w01 = W0e[1 * NH + t];
    const float w02 = W0e[2 * NH + t];
    const float cv  = c0[e * NH + t];
#pragma unroll 4
    for (int r = 0; r < TM; ++r) {
      const float v = fmaf(xs[r * 4 + 2], w02,
                      fmaf(xs[r * 4 + 1], w01,
                      fmaf(xs[r * 4 + 0], w00, cv)));
      hA[r * LDH + t] = fmaxf(v, 0.0f);
    }
  }
  __syncthreads();

  // ---- layers 1 & 2: 64x256x256 WMMA GEMMs
  gemm_relu(hA, hB, W1 + (size_t)e * NH * NH, b1 + e * NH);
  __syncthreads();
  gemm_relu(hB, hA, W2 + (size_t)e * NH * NH, b2 + e * NH);
  __syncthreads();

  // ---- head: out = h2 @ W3 + b3 (dot per row, 4-way split reduction)
  {
    const int row = t >> 2;
    const int q   = t & 3;
    const int cb  = q * 64;
    float s = 0.0f;
#pragma unroll 8
    for (int c = 0; c < 64; ++c)
      s = fmaf(hA[row * LDH + cb + c], w3s[cb + c], s);
    red[row * 4 + q] = s;
  }
  __syncthreads();
  if (t < TM) {
    const int i = ridx[t];
    if (i >= 0)
      out[i] = red[t * 4 + 0] + red[t * 4 + 1] + red[t * 4 + 2] + red[t * 4 + 3]
             + b3[e];
  }
}

// ------------------------------ launcher -----------------------------------

extern "C" void kernel_launch(void* const* d_in, const int* in_sizes, int n_in,
                              void* d_out, int out_size, void* d_ws, size_t ws_size,
                              hipStream_t stream) {
  const float* x      = (const float*)d_in[0];
  const int*   labels = (const int*)  d_in[1];
  const float* emb    = (const float*)d_in[2];
  const float* W0     = (const float*)d_in[3];
  const float* b0     = (const float*)d_in[4];
  const float* W1     = (const float*)d_in[5];
  const float* b1     = (const float*)d_in[6];
  const float* W2     = (const float*)d_in[7];
  const float* b2     = (const float*)d_in[8];
  const float* W3     = (const float*)d_in[9];
  const float* b3     = (const float*)d_in[10];
  float*       out    = (float*)d_out;

  int*   meta = (int*)d_ws;            // M_SIZE ints
  int*   perm = meta + M_SIZE;         // NPTS ints
  float* c0   = (float*)(perm + NPTS); // NE*NH floats

  zero_meta_kernel<<<1, 64, 0, stream>>>(meta);
  count_kernel<<<NPTS / 256, 256, 0, stream>>>(labels, meta);
  scan_kernel<<<1, 32, 0, stream>>>(meta);
  scatter_kernel<<<NPTS / 256, 256, 0, stream>>>(labels, meta, perm);
  c0_kernel<<<NE, NH, 0, stream>>>(emb, W0, b0, c0);

  const int maxTiles = NPTS / TM + NE;  // >= total tiles incl. per-expert padding
  moe_forward_kernel<<<maxTiles, 256, SMEM_BYTES, stream>>>(
      x, W0, W1, b1, W2, b2, W3, b3, meta, perm, c0, out);
}